// MapperSetsAE_49778670960667
// MI455X (gfx1250) — compile-verified
//
#include <hip/hip_runtime.h>
#include <math.h>

typedef __attribute__((ext_vector_type(2))) float v2f;
typedef __attribute__((ext_vector_type(8))) float v8f;

constexpr int B = 64, N = 128, D = 2048;

// ---------------------------------------------------------------------------
// Kernel 1: squared row norms for X (grid.y==0) and Y (grid.y==1)
// ---------------------------------------------------------------------------
__global__ __launch_bounds__(256) void row_norms_kernel(
    const float* __restrict__ X, const float* __restrict__ Y,
    float* __restrict__ x2, float* __restrict__ y2) {
  const float* src = (blockIdx.y == 0 ? X : Y) + (size_t)blockIdx.x * D;
  float*       dst = (blockIdx.y == 0 ? x2 : y2);
  float s = 0.f;
  for (int k = threadIdx.x * 4; k < D; k += 256 * 4) {
    float4 v = *(const float4*)(src + k);
    s += v.x * v.x + v.y * v.y + v.z * v.z + v.w * v.w;
  }
  __shared__ float red[256];
  red[threadIdx.x] = s;
  __syncthreads();
  for (int off = 128; off > 0; off >>= 1) {
    if (threadIdx.x < (unsigned)off) red[threadIdx.x] += red[threadIdx.x + off];
    __syncthreads();
  }
  if (threadIdx.x == 0) dst[blockIdx.x] = red[0];
}

// ---------------------------------------------------------------------------
// Kernel 2: cost[b][m][n] = sqrt(max(x2[m] + y2[n] - 2*dot(X[m],Y[n]), 0))
// One wave32 per 16x32 tile: A fragment loaded once per K-step and reused
// across two N-tiles (two accumulators) -> 3 loads per 2 WMMAs.
// fp32 WMMA 16x16x4; A 16x4 layout (ISA 7.12.2): lanes 0-15 hold K={k,k+1},
// lanes 16-31 hold K={k+2,k+3} for row M = lane&15; B 4x16 mirrors it with
// column N = lane&15.
// ---------------------------------------------------------------------------
__global__ __launch_bounds__(32) void cost_wmma_kernel(
    const float* __restrict__ X, const float* __restrict__ Y,
    const float* __restrict__ x2, const float* __restrict__ y2,
    float* __restrict__ cost) {
  const int b    = blockIdx.z;
  const int tm   = blockIdx.y * 16;
  const int tn   = blockIdx.x * 32;
  const int lane = threadIdx.x;
  const int half = lane >> 4;   // 0: K pair {0,1}, 1: K pair {2,3}
  const int l    = lane & 15;

  const float* ap  = X + ((size_t)b * N + (tm + l)) * D + half * 2;
  const float* bp0 = Y + ((size_t)b * N + (tn + l)) * D + half * 2;
  const float* bp1 = bp0 + (size_t)16 * D;

  v8f acc0 = {};
  v8f acc1 = {};
#pragma unroll 8
  for (int k = 0; k < D; k += 4) {
    v2f a  = *(const v2f*)(ap + k);
    v2f b0 = *(const v2f*)(bp0 + k);
    v2f b1 = *(const v2f*)(bp1 + k);
    acc0 = __builtin_amdgcn_wmma_f32_16x16x4_f32(
        false, a, false, b0, (short)0, acc0, false, false);
    acc1 = __builtin_amdgcn_wmma_f32_16x16x4_f32(
        false, a, false, b1, (short)0, acc1, false, false);
  }

  // C layout: VGPR r holds (M = tm + r + 8*half, N = tile_n + (lane&15))
  const float yv0  = y2[b * N + tn + l];
  const float yv1  = y2[b * N + tn + 16 + l];
  float*      crow = cost + ((size_t)b * N) * N;
#pragma unroll
  for (int r = 0; r < 8; ++r) {
    const int   m  = tm + r + 8 * half;
    const float xv = x2[b * N + m];
    const float d0 = xv + yv0 - 2.0f * acc0[r];
    const float d1 = xv + yv1 - 2.0f * acc1[r];
    crow[(size_t)m * N + (tn + l)]      = sqrtf(fmaxf(d0, 0.0f));
    crow[(size_t)m * N + (tn + 16 + l)] = sqrtf(fmaxf(d1, 0.0f));
  }
}

// ---------------------------------------------------------------------------
// Kernel 3: Jonker-Volgenant assignment, one wave32 per batch.
// Columns j=1..128 live 4-per-lane in registers (j = 1 + lane + 32*s);
// u[], p[] in LDS (row-indexed / path state). float64 duals like the reference.
// ---------------------------------------------------------------------------
__global__ __launch_bounds__(32) void hungarian_kernel(
    const float* __restrict__ cost, int* __restrict__ cols) {
  const int b    = blockIdx.x;
  const int lane = threadIdx.x;
  const float* C = cost + (size_t)b * N * N;

  __shared__ double u[N + 1];
  __shared__ int    p[N + 1];

  double v_s[4]    = {0.0, 0.0, 0.0, 0.0};
  double minv_s[4];
  int    way_s[4]  = {0, 0, 0, 0};

  for (int j = lane; j <= N; j += 32) { u[j] = 0.0; p[j] = 0; }
  __syncthreads();

  const double INF = 1e300;

  for (int i = 1; i <= N; ++i) {
    if (lane == 0) p[0] = i;
    __syncthreads();
    int      j0       = 0;
    unsigned usedmask = 0;   // bit s -> column j = 1 + lane + 32*s is used
#pragma unroll
    for (int s = 0; s < 4; ++s) minv_s[s] = INF;

    while (true) {
      if (j0 > 0 && ((j0 - 1) & 31) == lane) usedmask |= 1u << ((j0 - 1) >> 5);
      const int    i0   = p[j0];
      const double ui0  = u[i0];
      const float* crow = C + (size_t)(i0 - 1) * N;
#pragma unroll
      for (int s = 0; s < 4; ++s) {
        if (!((usedmask >> s) & 1u)) {
          const double cand = (double)crow[lane + 32 * s] - ui0 - v_s[s];
          if (cand < minv_s[s]) { minv_s[s] = cand; way_s[s] = j0; }
        }
      }
      // argmin over free columns; ties -> smallest j (np.argmin semantics)
      double best  = INF;
      int    bestj = 0x7fffffff;
#pragma unroll
      for (int s = 0; s < 4; ++s) {
        if (!((usedmask >> s) & 1u)) {
          const int j = 1 + lane + 32 * s;
          if (minv_s[s] < best || (minv_s[s] == best && j < bestj)) {
            best = minv_s[s]; bestj = j;
          }
        }
      }
      for (int off = 16; off > 0; off >>= 1) {
        const double ob = __shfl_xor(best, off, 32);
        const int    oj = __shfl_xor(bestj, off, 32);
        if (ob < best || (ob == best && oj < bestj)) { best = ob; bestj = oj; }
      }
      const double delta = best;
      const int    j1    = bestj;

      if (lane == 0) u[p[0]] += delta;  // j=0 is always "used"
#pragma unroll
      for (int s = 0; s < 4; ++s) {
        if ((usedmask >> s) & 1u) {
          const int j = 1 + lane + 32 * s;
          u[p[j]] += delta;             // distinct rows per used column
          v_s[s]  -= delta;
        } else {
          minv_s[s] -= delta;
        }
      }
      __syncthreads();
      j0 = j1;
      if (p[j0] == 0) break;
    }
    // augment along alternating path (way lives in lane registers)
    while (j0 != 0) {
      const int jm1 = j0 - 1;
      const int s   = jm1 >> 5;
      const int src = jm1 & 31;
      const int w0  = __shfl(way_s[0], src, 32);
      const int w1  = __shfl(way_s[1], src, 32);
      const int w2  = __shfl(way_s[2], src, 32);
      const int w3  = __shfl(way_s[3], src, 32);
      const int j1  = (s == 0) ? w0 : (s == 1) ? w1 : (s == 2) ? w2 : w3;
      if (lane == 0) p[j0] = p[j1];
      __syncthreads();
      j0 = j1;
    }
    __syncthreads();
  }
  // cols[p[j]-1] = j-1
#pragma unroll
  for (int s = 0; s < 4; ++s) {
    const int j = 1 + lane + 32 * s;
    cols[b * N + (p[j] - 1)] = j - 1;
  }
}

// ---------------------------------------------------------------------------
// Kernel 4: gather mapped outputs / probs / cols into d_out.
// d_out layout: [B*N*D] mapped_outputs, then [B*N] mapped_prob, then [B*N] cols
// ---------------------------------------------------------------------------
__global__ __launch_bounds__(256) void gather_kernel(
    const float* __restrict__ Y, const float* __restrict__ prob,
    const int* __restrict__ cols, float* __restrict__ out) {
  const int idx = blockIdx.x;    // b*N + i
  const int b   = idx >> 7;
  const int c   = cols[idx];
  const float* src = Y + ((size_t)b * N + c) * D;
  float*       dst = out + (size_t)idx * D;
  for (int k = threadIdx.x * 4; k < D; k += 256 * 4) {
    *(float4*)(dst + k) = *(const float4*)(src + k);
  }
  if (threadIdx.x == 0) {
    out[(size_t)B * N * D + idx]          = prob[b * N + c];
    out[(size_t)B * N * D + B * N + idx]  = (float)c;
  }
}

// ---------------------------------------------------------------------------
extern "C" void kernel_launch(void* const* d_in, const int* in_sizes, int n_in,
                              void* d_out, int out_size, void* d_ws, size_t ws_size,
                              hipStream_t stream) {
  const float* X = (const float*)d_in[0];  // inputs               [B,N,D]
  const float* Y = (const float*)d_in[1];  // stacked_outputs      [B,N,D]
  const float* P = (const float*)d_in[2];  // member_probabilities [B,N,1]
  float*       out = (float*)d_out;

  float* cost = (float*)d_ws;                       // B*N*N floats (4 MB)
  float* x2   = cost + (size_t)B * N * N;           // B*N floats
  float* y2   = x2 + B * N;                         // B*N floats
  int*   cols = (int*)(y2 + B * N);                 // B*N ints

  row_norms_kernel<<<dim3(B * N, 2), 256, 0, stream>>>(X, Y, x2, y2);
  cost_wmma_kernel<<<dim3(4, 8, B), 32, 0, stream>>>(X, Y, x2, y2, cost);
  hungarian_kernel<<<B, 32, 0, stream>>>(cost, cols);
  gather_kernel<<<B * N, 256, 0, stream>>>(Y, P, cols, out);
}